// MoERouter_22385369547513
// MI455X (gfx1250) — compile-verified
//
#include <hip/hip_runtime.h>
#include <hip/hip_bf16.h>

typedef float v2f __attribute__((ext_vector_type(2)));
typedef float v8f __attribute__((ext_vector_type(8)));

#define D_MODEL 1024
#define NEXP    64
#define KC      128      // K chunk staged in LDS per iteration
#define LDA     132      // LDS row pitch for A (16B aligned rows, conflict-free)
#define LDW     132      // LDS row pitch for W
#define LDL     65       // LDS pitch for logits (bank-conflict-free column scan)

__global__ __launch_bounds__(128)
void moe_router_fused(const float* __restrict__ x,
                      const float* __restrict__ W,
                      const int*   __restrict__ xmask,
                      float* __restrict__ out_w,
                      float* __restrict__ out_i,
                      float* __restrict__ out_l,
                      float* __restrict__ out_p)
{
    __shared__ float sA[16 * LDA];        // 16 tokens x KC
    __shared__ float sW[NEXP * LDW];      // 64 experts x KC
    __shared__ float sL[16 * LDL];        // logits tile
    __shared__ int   sMask[16];
    __shared__ float sMax[16], sSum[16];

    const int tid  = threadIdx.x;
    const int lane = tid & 31;
    const int wave = tid >> 5;            // 0..3, expert group
    const int m0   = blockIdx.x * 16;     // first token of this tile

    if (tid < 16) sMask[tid] = xmask[m0 + tid];

    const int row16 = lane & 15;
    const int khalf = (lane >> 4) << 1;   // 0 or 2: K-pair this half-wave owns

    // cooperative-load mappings
    const int arow = tid >> 3;            // 8 threads per A row, 16 floats each
    const int acol = (tid & 7) * 16;
    const int wrow = tid >> 1;            // 2 threads per W row, 64 floats each
    const int wcol = (tid & 1) * 64;

    v8f c = {};

    for (int k0 = 0; k0 < D_MODEL; k0 += KC) {
        // stage A chunk (16 x 128 f32), fully coalesced b128 loads
        {
            const float4* gx = (const float4*)(x + (size_t)(m0 + arow) * D_MODEL + k0 + acol);
            float4* lA = (float4*)(sA + arow * LDA + acol);
#pragma unroll
            for (int j = 0; j < 4; ++j) lA[j] = gx[j];
        }
        // stage W chunk (64 x 128 f32) — W is L2-resident, cheap
        {
            const float4* gw = (const float4*)(W + (size_t)wrow * D_MODEL + k0 + wcol);
            float4* lW = (float4*)(sW + wrow * LDW + wcol);
#pragma unroll
            for (int j = 0; j < 16; ++j) lW[j] = gw[j];
        }
        // prefetch next x chunk while we compute
        if (k0 + KC < D_MODEL)
            __builtin_prefetch(x + (size_t)(m0 + arow) * D_MODEL + k0 + KC + acol, 0, 0);
        __syncthreads();

        const float* pa = sA + row16 * LDA + khalf;
        const float* pb = sW + (wave * 16 + row16) * LDW + khalf;
#pragma unroll
        for (int kk = 0; kk < KC; kk += 4) {
            v2f a = *(const v2f*)(pa + kk);
            v2f b = *(const v2f*)(pb + kk);
            // D = A(16x4 f32) * B(4x16 f32) + C
            c = __builtin_amdgcn_wmma_f32_16x16x4_f32(
                    false, a, false, b, (short)0, c, false, false);
        }
        __syncthreads();
    }

    // ---- epilogue: logits -> LDS + masked logits -> global ----
    const int rbase = (lane >> 4) * 8;           // C/D layout: lanes>=16 hold rows 8..15
    const int nE    = wave * 16 + row16;         // expert column
#pragma unroll
    for (int v = 0; v < 8; ++v) {
        const int r  = rbase + v;
        const float lg = c[v];
        sL[r * LDL + nE] = lg;
        out_l[(size_t)(m0 + r) * NEXP + nE] = sMask[r] ? lg : 0.0f;
    }
    __syncthreads();

    // ---- softmax stats + top-2 per token (16 threads) ----
    if (tid < 16) {
        const int r = tid;
        const float* lrow = sL + r * LDL;
        float mx = lrow[0];
#pragma unroll 8
        for (int e = 1; e < NEXP; ++e) mx = fmaxf(mx, lrow[e]);
        float sum = 0.0f;
        float v0 = -1.0f, v1 = -1.0f;
        int   i0 = 0,     i1 = 0;
        for (int e = 0; e < NEXP; ++e) {
            const float p = __expf(lrow[e] - mx);   // monotonic with probs
            sum += p;
            if (p > v0)      { v1 = v0; i1 = i0; v0 = p; i0 = e; }
            else if (p > v1) { v1 = p;  i1 = e; }
        }
        sMax[r] = mx;
        sSum[r] = sum;
        const int   mk    = sMask[r];
        const float denom = v0 + v1;                // p0/(p0+p1) == e0/(e0+e1)
        const size_t t    = (size_t)(m0 + r);
        out_w[t * 2 + 0] = mk ? (v0 / denom) : 0.0f;
        out_w[t * 2 + 1] = mk ? (v1 / denom) : 0.0f;
        out_i[t * 2 + 0] = mk ? (float)i0 : -1.0f;
        out_i[t * 2 + 1] = mk ? (float)i1 : -1.0f;
    }
    __syncthreads();

    // ---- masked probs -> global (all 128 threads, contiguous 8-float spans) ----
    {
        const int r  = tid >> 3;
        const int e0 = (tid & 7) * 8;
        const float mx  = sMax[r];
        const float inv = 1.0f / sSum[r];
        const float mkf = sMask[r] ? 1.0f : 0.0f;
        float* po = out_p + (size_t)(m0 + r) * NEXP + e0;
#pragma unroll
        for (int j = 0; j < 8; ++j)
            po[j] = __expf(sL[r * LDL + e0 + j] - mx) * inv * mkf;
    }
}

extern "C" void kernel_launch(void* const* d_in, const int* in_sizes, int n_in,
                              void* d_out, int out_size, void* d_ws, size_t ws_size,
                              hipStream_t stream) {
    const float* x     = (const float*)d_in[0];   // (B,T,D) f32
    const float* W     = (const float*)d_in[1];   // (E,D)   f32
    const int*   xmask = (const int*)  d_in[2];   // (B,T)   i32

    const int BT = in_sizes[2];                   // B*T = 32768

    float* out   = (float*)d_out;
    float* out_w = out;                           // (BT,2) expert_weights
    float* out_i = out + (size_t)BT * 2;          // (BT,2) expert_indices (as float)
    float* out_l = out + (size_t)BT * 4;          // (BT,64) router_logits
    float* out_p = out_l + (size_t)BT * NEXP;     // (BT,64) router_probs

    const int tiles = BT / 16;                    // 2048 blocks
    moe_router_fused<<<tiles, 128, 0, stream>>>(x, W, xmask, out_w, out_i, out_l, out_p);
}